// OptimizedDistance_75376676045589
// MI455X (gfx1250) — compile-verified
//
#include <hip/hip_runtime.h>
#include <hip/hip_bf16.h>

#define N_ATOMS   4096
#define MAX_NB    128
#define MAX_PAIRS (MAX_NB * N_ATOMS)   // 524288
#define CUT2      25.0f
#define ITILES    (N_ATOMS / 16)       // 256
#define WAVES_PER_BLOCK 4

typedef __attribute__((ext_vector_type(2))) float v2f;
typedef __attribute__((ext_vector_type(8))) float v8f;

// Shared fixup so count + fill predicates are bit-identical:
// d2 = |ri|^2 + |rj|^2 - 2*(ri.rj)   with the dot product from WMMA.
static __device__ __forceinline__ float nl_d2(float g, float rni, float rnj) {
    return __builtin_fmaf(-2.0f, g, rni + rnj);
}

// Combined per-k acceptance mask for one 16x16 tile (identical in both passes):
//   ballot(d2 < cut2) & ballot(batch match) & diagonal-exclusion mask.
// All three terms are SGPR masks when `diag` is scalar, so this is pure SALU.
// Diagonal tile (jt == wave): row k excludes lane k (low half, col==k) and
// row 8+k excludes lane 24+k (high half, col==8+k).
static __device__ __forceinline__ unsigned nl_mask(float d2, bool bmatch,
                                                   bool diag, int k) {
    unsigned m = __builtin_amdgcn_ballot_w32(d2 < CUT2)
               & __builtin_amdgcn_ballot_w32(bmatch);
    unsigned self = diag ? ~((1u << k) | (1u << (24 + k))) : 0xFFFFFFFFu;
    return m & self;
}

// ---------------------------------------------------------------------------
// Pass 1: per-row neighbor counts. One wave owns a 16-row i-tile and streams
// all 256 16-column j-tiles. The 16x16 dot-product tile comes from
// V_WMMA_F32_16X16X4_F32 (A = 16x4 [x y z 0], B = 4x16).
// ---------------------------------------------------------------------------
__global__ __launch_bounds__(32 * WAVES_PER_BLOCK)
void nl_count_kernel(const float* __restrict__ pos,
                     const int*   __restrict__ batch,
                     int*         __restrict__ counts) {
    const int lane    = threadIdx.x & 31;
    // provably wave-uniform -> keeps diag masks / popcounts / accums in SALU
    const int wave    = __builtin_amdgcn_readfirstlane(
                            blockIdx.x * WAVES_PER_BLOCK + (threadIdx.x >> 5));
    const int ibase   = wave * 16;
    const int col     = lane & 15;
    const bool hihalf = lane >= 16;
    const int rowbase = hihalf ? 8 : 0;

    // A-tile (16x4, K = x,y,z,0): lanes 0-15 hold K0/K1, lanes 16-31 hold K2/K3
    float ax = pos[(ibase + col) * 3 + 0];
    float ay = pos[(ibase + col) * 3 + 1];
    float az = pos[(ibase + col) * 3 + 2];
    v2f a;
    a.x = hihalf ? az : ax;
    a.y = hihalf ? 0.0f : ay;

    // Per-lane data for this half's 8 rows (C/D layout: VGPR k = rows k, k+8)
    float rni[8];
    int   bi[8];
    #pragma unroll
    for (int k = 0; k < 8; ++k) {
        int i = ibase + rowbase + k;
        float x = pos[i * 3 + 0], y = pos[i * 3 + 1], z = pos[i * 3 + 2];
        rni[k] = x * x + y * y + z * z;
        bi[k]  = batch[i];
    }

    int rcLo[8], rcHi[8];
    #pragma unroll
    for (int k = 0; k < 8; ++k) { rcLo[k] = 0; rcHi[k] = 0; }

    for (int jt = 0; jt < ITILES; ++jt) {
        const int j = jt * 16 + col;
        float qx = pos[j * 3 + 0];
        float qy = pos[j * 3 + 1];
        float qz = pos[j * 3 + 2];
        float rnj = qx * qx + qy * qy + qz * qz;
        int   bj  = batch[j];
        const bool diag = (jt == wave);

        v2f b;
        b.x = hihalf ? qz : qx;
        b.y = hihalf ? 0.0f : qy;

        v8f c = {};
        c = __builtin_amdgcn_wmma_f32_16x16x4_f32(
                false, a, false, b, (short)0, c, false, false);

        #pragma unroll
        for (int k = 0; k < 8; ++k) {
            float d2   = nl_d2(c[k], rni[k], rnj);
            unsigned m = nl_mask(d2, bi[k] == bj, diag, k);
            rcLo[k] += __popc(m & 0xFFFFu);
            rcHi[k] += __popc(m >> 16);
        }
    }

    if (lane == 0) {
        #pragma unroll
        for (int k = 0; k < 8; ++k) {
            counts[ibase + k]     = rcLo[k];
            counts[ibase + 8 + k] = rcHi[k];
        }
    }
}

// ---------------------------------------------------------------------------
// Pass 2: exclusive scan of the 4096 row counts (single block).
// offsets[r] = start slot of row r, offsets[4096] = total pair count.
// ---------------------------------------------------------------------------
__global__ __launch_bounds__(1024)
void nl_scan_kernel(const int* __restrict__ counts, int* __restrict__ offsets) {
    __shared__ int sums[1024];
    const int t = threadIdx.x;
    int c0 = counts[4 * t + 0];
    int c1 = counts[4 * t + 1];
    int c2 = counts[4 * t + 2];
    int c3 = counts[4 * t + 3];
    int s  = c0 + c1 + c2 + c3;
    sums[t] = s;
    __syncthreads();
    for (int off = 1; off < 1024; off <<= 1) {
        int add = (t >= off) ? sums[t - off] : 0;
        __syncthreads();
        sums[t] += add;
        __syncthreads();
    }
    int excl = sums[t] - s;
    offsets[4 * t + 0] = excl;
    offsets[4 * t + 1] = excl + c0;
    offsets[4 * t + 2] = excl + c0 + c1;
    offsets[4 * t + 3] = excl + c0 + c1 + c2;
    if (t == 1023) offsets[4096] = sums[t];
}

// ---------------------------------------------------------------------------
// Pass 3: ordered fill. Same WMMA + mask expression as pass 1; within-row
// order is ascending j (prefix popcount of the mask), rows start at their
// scanned offsets — exactly the reference's row-major compaction.
// ---------------------------------------------------------------------------
__global__ __launch_bounds__(32 * WAVES_PER_BLOCK)
void nl_pair_kernel(const float* __restrict__ pos,
                    const int*   __restrict__ batch,
                    const int*   __restrict__ offsets,
                    int*   __restrict__ ei,
                    int*   __restrict__ ej,
                    float* __restrict__ ew,
                    float* __restrict__ ev) {
    const int lane    = threadIdx.x & 31;
    const int wave    = __builtin_amdgcn_readfirstlane(
                            blockIdx.x * WAVES_PER_BLOCK + (threadIdx.x >> 5));
    const int ibase   = wave * 16;
    const int col     = lane & 15;
    const bool hihalf = lane >= 16;
    const int rowbase = hihalf ? 8 : 0;

    float ax = pos[(ibase + col) * 3 + 0];
    float ay = pos[(ibase + col) * 3 + 1];
    float az = pos[(ibase + col) * 3 + 2];
    v2f a;
    a.x = hihalf ? az : ax;
    a.y = hihalf ? 0.0f : ay;

    float xi[8], yi[8], zi[8], rni[8];
    int   bi[8];
    int   rpLo[8], rpHi[8];
    #pragma unroll
    for (int k = 0; k < 8; ++k) {
        int i = ibase + rowbase + k;
        xi[k] = pos[i * 3 + 0];
        yi[k] = pos[i * 3 + 1];
        zi[k] = pos[i * 3 + 2];
        rni[k] = xi[k] * xi[k] + yi[k] * yi[k] + zi[k] * zi[k];
        bi[k]  = batch[i];
        rpLo[k] = offsets[ibase + k];
        rpHi[k] = offsets[ibase + 8 + k];
    }

    for (int jt = 0; jt < ITILES; ++jt) {
        const int j = jt * 16 + col;
        float qx = pos[j * 3 + 0];
        float qy = pos[j * 3 + 1];
        float qz = pos[j * 3 + 2];
        float rnj = qx * qx + qy * qy + qz * qz;
        int   bj  = batch[j];
        const bool diag = (jt == wave);

        v2f b;
        b.x = hihalf ? qz : qx;
        b.y = hihalf ? 0.0f : qy;

        v8f c = {};
        c = __builtin_amdgcn_wmma_f32_16x16x4_f32(
                false, a, false, b, (short)0, c, false, false);

        #pragma unroll
        for (int k = 0; k < 8; ++k) {
            float d2     = nl_d2(c[k], rni[k], rnj);
            unsigned m   = nl_mask(d2, bi[k] == bj, diag, k);
            unsigned mlo = m & 0xFFFFu;
            unsigned mhi = m >> 16;
            if (m & (1u << lane)) {
                unsigned half = hihalf ? mhi : mlo;
                int prefix = __popc(half & ((1u << col) - 1u));
                int slot   = (hihalf ? rpHi[k] : rpLo[k]) + prefix;
                if (slot < MAX_PAIRS) {
                    // exact reference values recomputed from positions
                    float vx = xi[k] - qx;
                    float vy = yi[k] - qy;
                    float vz = zi[k] - qz;
                    float dd = vx * vx + vy * vy + vz * vz;
                    ei[slot] = ibase + rowbase + k;
                    ej[slot] = j;
                    ew[slot] = __builtin_sqrtf(dd);
                    ev[3 * slot + 0] = vx;
                    ev[3 * slot + 1] = vy;
                    ev[3 * slot + 2] = vz;
                }
            }
            rpLo[k] += __popc(mlo);
            rpHi[k] += __popc(mhi);
        }
    }
}

// ---------------------------------------------------------------------------
// Pass 4: pad only the unused tail with (-1, 0) — live slots already written.
// ---------------------------------------------------------------------------
__global__ __launch_bounds__(256)
void nl_pad_kernel(const int* __restrict__ offsets,
                   int*   __restrict__ ei,
                   int*   __restrict__ ej,
                   float* __restrict__ ew,
                   float* __restrict__ ev) {
    int idx = blockIdx.x * blockDim.x + threadIdx.x;
    if (idx >= MAX_PAIRS) return;
    int total = offsets[N_ATOMS];
    if (total > MAX_PAIRS) total = MAX_PAIRS;
    if (idx >= total) {
        ei[idx] = -1;
        ej[idx] = -1;
        ew[idx] = 0.0f;
        ev[3 * idx + 0] = 0.0f;
        ev[3 * idx + 1] = 0.0f;
        ev[3 * idx + 2] = 0.0f;
    }
}

extern "C" void kernel_launch(void* const* d_in, const int* in_sizes, int n_in,
                              void* d_out, int out_size, void* d_ws, size_t ws_size,
                              hipStream_t stream) {
    const float* pos   = (const float*)d_in[0];   // (N, 3) f32
    const int*   batch = (const int*)d_in[1];     // (N,)  i32

    // workspace: counts[4096] | offsets[4097]
    int* counts  = (int*)d_ws;
    int* offsets = counts + N_ATOMS;

    // d_out: edge_i[MP] | edge_j[MP] (int32 bits) | weight[MP] | vec[MP*3]
    int*   ei = (int*)d_out;
    int*   ej = ei + MAX_PAIRS;
    float* ew = (float*)d_out + 2 * MAX_PAIRS;
    float* ev = ew + MAX_PAIRS;

    const int blocks = ITILES / WAVES_PER_BLOCK;  // 64 blocks x 4 waves
    nl_count_kernel<<<blocks, 32 * WAVES_PER_BLOCK, 0, stream>>>(pos, batch, counts);
    nl_scan_kernel<<<1, 1024, 0, stream>>>(counts, offsets);
    nl_pair_kernel<<<blocks, 32 * WAVES_PER_BLOCK, 0, stream>>>(pos, batch, offsets,
                                                                ei, ej, ew, ev);
    nl_pad_kernel<<<(MAX_PAIRS + 255) / 256, 256, 0, stream>>>(offsets, ei, ej, ew, ev);
}